// GATLayerOut_single_pyg_1932735283951
// MI455X (gfx1250) — compile-verified
//
#include <hip/hip_runtime.h>
#include <hip/hip_bf16.h>

typedef __attribute__((ext_vector_type(2))) float v2f;
typedef __attribute__((ext_vector_type(8))) float v8f;

#define DF 128
#define NEG_SLOPE 0.2f

// ---------------------------------------------------------------------------
// Init: zero accumulator (d_out), mmax = -inf, denom = 0
// ---------------------------------------------------------------------------
__global__ void init_kernel(float* __restrict__ out, float* __restrict__ mmax,
                            float* __restrict__ denom, int N) {
  int i = blockIdx.x * blockDim.x + threadIdx.x;
  int total = N * DF;
  if (i < total) out[i] = 0.0f;
  if (i < N) {
    mmax[i]  = -__builtin_inff();
    denom[i] = 0.0f;
  }
}

// ---------------------------------------------------------------------------
// GEMM: xw = x @ W via V_WMMA_F32_16X16X4_F32 (fp32 in/out).
// One wave per 16-row strip of x; 8 accumulators cover all of DOUT=128,
// so each A fragment is loaded once and reused for 8 WMMAs (8x less A
// traffic than tile-per-wave). K loop in steps of 4.
// A 16x4 layout: lanes 0-15 -> (K=k,k+1), lanes 16-31 -> (K=k+2,k+3).
// B 4x16 layout (symmetric): lane&15 -> column, lane half selects K pair.
// C/D: 8 VGPRs; lane half selects M = v or v+8.
// ---------------------------------------------------------------------------
__global__ void gemm_xw_kernel(const float* __restrict__ x,
                               const float* __restrict__ W,
                               float* __restrict__ xw, int N) {
  int wave  = threadIdx.x >> 5;
  int lane  = threadIdx.x & 31;
  int strip = blockIdx.x * (blockDim.x >> 5) + wave;
  int tilesM = (N + 15) >> 4;
  if (strip >= tilesM) return;          // uniform per wave

  int half = lane >> 4;                 // 0: K pair (k,k+1); 1: (k+2,k+3)
  int l    = lane & 15;

  int rowA = strip * 16 + l;
  if (rowA >= N) rowA = N - 1;          // never triggers when N % 16 == 0
  const float* Arow = x + (size_t)rowA * DF;

  v8f zero = {};
  v8f acc[8];
  #pragma unroll
  for (int t = 0; t < 8; ++t) acc[t] = zero;

  for (int k = 0; k < DF; k += 4) {
    int ka = k + 2 * half;
    v2f a;
    a.x = Arow[ka];
    a.y = Arow[ka + 1];
    const float* Wrow = W + (size_t)ka * DF + l;
    #pragma unroll
    for (int t = 0; t < 8; ++t) {
      v2f b;
      b.x = Wrow[t * 16];
      b.y = Wrow[DF + t * 16];
      acc[t] = __builtin_amdgcn_wmma_f32_16x16x4_f32(
          /*neg_a=*/false, a, /*neg_b=*/false, b,
          /*c_mod=*/(short)0, acc[t], /*reuse_a=*/false, /*reuse_b=*/false);
    }
  }

  int mbase = strip * 16 + half * 8;
  if (strip * 16 + 16 <= N) {           // fast path: full strip (always here)
    #pragma unroll
    for (int t = 0; t < 8; ++t) {
      float* orow = xw + (size_t)mbase * DF + t * 16 + l;
      #pragma unroll
      for (int v = 0; v < 8; ++v) orow[(size_t)v * DF] = acc[t][v];
    }
  } else {                              // guarded tail (dead for N % 16 == 0)
    for (int t = 0; t < 8; ++t)
      for (int v = 0; v < 8; ++v) {
        int row = mbase + v;
        if (row < N) xw[(size_t)row * DF + t * 16 + l] = acc[t][v];
      }
  }
}

// ---------------------------------------------------------------------------
// Per-node attention logits: a_src[n] = xw[n,:].att_src ; a_dst likewise.
// One wave per node, float4 per lane, shfl reduction (wave32).
// ---------------------------------------------------------------------------
__global__ void logits_kernel(const float* __restrict__ xw,
                              const float* __restrict__ att_src,
                              const float* __restrict__ att_dst,
                              float* __restrict__ a_src,
                              float* __restrict__ a_dst, int N) {
  int gw   = (blockIdx.x * blockDim.x + threadIdx.x) >> 5;
  int lane = threadIdx.x & 31;
  if (gw >= N) return;                  // uniform per wave
  const float4 xv = *(const float4*)(xw + (size_t)gw * DF + lane * 4);
  const float4 sv = *(const float4*)(att_src + lane * 4);
  const float4 dv = *(const float4*)(att_dst + lane * 4);
  float s = xv.x * sv.x + xv.y * sv.y + xv.z * sv.z + xv.w * sv.w;
  float d = xv.x * dv.x + xv.y * dv.y + xv.z * dv.z + xv.w * dv.w;
  #pragma unroll
  for (int off = 16; off > 0; off >>= 1) {
    s += __shfl_down(s, off, 32);
    d += __shfl_down(d, off, 32);
  }
  if (lane == 0) { a_src[gw] = s; a_dst[gw] = d; }
}

// ---------------------------------------------------------------------------
// Edge helpers
// ---------------------------------------------------------------------------
__device__ __forceinline__ void edge_sd(const int* __restrict__ ei, int i,
                                        int E, int& s, int& d) {
  if (i < E) { s = ei[i]; d = ei[E + i]; }
  else       { s = i - E; d = s; }      // self loop
}

__device__ __forceinline__ float leaky(float e) {
  return e >= 0.0f ? e : NEG_SLOPE * e;
}

__device__ __forceinline__ void atomicMaxFloat(float* addr, float val) {
  if (val >= 0.0f) atomicMax((int*)addr, __float_as_int(val));
  else atomicMin((unsigned int*)addr, (unsigned int)__float_as_int(val));
}

// Pass 1: segment max over destinations
__global__ void edge_max_kernel(const int* __restrict__ ei,
                                const float* __restrict__ a_src,
                                const float* __restrict__ a_dst,
                                float* __restrict__ mmax, int E, int N) {
  int i = blockIdx.x * blockDim.x + threadIdx.x;
  if (i >= E + N) return;
  int s, d; edge_sd(ei, i, E, s, d);
  float e = leaky(a_src[s] + a_dst[d]);
  atomicMaxFloat(&mmax[d], e);
}

// Pass 2: segment sum of exp(e - m[dst])
__global__ void edge_denom_kernel(const int* __restrict__ ei,
                                  const float* __restrict__ a_src,
                                  const float* __restrict__ a_dst,
                                  const float* __restrict__ mmax,
                                  float* __restrict__ denom, int E, int N) {
  int i = blockIdx.x * blockDim.x + threadIdx.x;
  if (i >= E + N) return;
  int s, d; edge_sd(ei, i, E, s, d);
  float e = leaky(a_src[s] + a_dst[d]);
  atomicAdd(&denom[d], expf(e - mmax[d]));
}

// Pass 3: out[dst] += alpha * xw[src]; one wave per edge, float4 per lane.
__global__ void edge_agg_kernel(const int* __restrict__ ei,
                                const float* __restrict__ xw,
                                const float* __restrict__ a_src,
                                const float* __restrict__ a_dst,
                                const float* __restrict__ mmax,
                                const float* __restrict__ denom,
                                float* __restrict__ out, int E, int N) {
  int gw   = (blockIdx.x * blockDim.x + threadIdx.x) >> 5;
  int lane = threadIdx.x & 31;
  if (gw >= E + N) return;              // uniform per wave
  int s, d; edge_sd(ei, gw, E, s, d);
  float e = leaky(a_src[s] + a_dst[d]);
  float alpha = expf(e - mmax[d]) / denom[d];
  const float4 xv = *(const float4*)(xw + (size_t)s * DF + lane * 4);
  float* o = out + (size_t)d * DF + lane * 4;
  atomicAdd(o + 0, xv.x * alpha);
  atomicAdd(o + 1, xv.y * alpha);
  atomicAdd(o + 2, xv.z * alpha);
  atomicAdd(o + 3, xv.w * alpha);
}

// Finalize: out = elu(out + bias)
__global__ void finalize_kernel(float* __restrict__ out,
                                const float* __restrict__ bias, int total) {
  int i = blockIdx.x * blockDim.x + threadIdx.x;
  if (i >= total) return;
  float v = out[i] + bias[i & (DF - 1)];
  out[i] = v > 0.0f ? v : (expf(v) - 1.0f);
}

// ---------------------------------------------------------------------------
extern "C" void kernel_launch(void* const* d_in, const int* in_sizes, int n_in,
                              void* d_out, int out_size, void* d_ws, size_t ws_size,
                              hipStream_t stream) {
  const float* x       = (const float*)d_in[0];
  const int*   ei      = (const int*)  d_in[1];
  const float* W       = (const float*)d_in[2];
  const float* att_src = (const float*)d_in[3];
  const float* att_dst = (const float*)d_in[4];
  const float* bias    = (const float*)d_in[5];

  const int N = in_sizes[0] / DF;       // 50000
  const int E = in_sizes[1] / 2;        // 800000
  float* out = (float*)d_out;

  float* xw    = (float*)d_ws;          // N*128
  float* asrc  = xw + (size_t)N * DF;   // N
  float* adst  = asrc + N;              // N
  float* mmax  = adst + N;              // N
  float* denom = mmax + N;              // N

  const int tot = E + N;

  // init accumulators
  {
    int threads = N * DF;
    init_kernel<<<(threads + 255) / 256, 256, 0, stream>>>(out, mmax, denom, N);
  }
  // GEMM (WMMA f32): one wave per 16-row strip, 4 waves per block
  {
    int strips = (N + 15) / 16;
    int blocks = (strips + 3) / 4;
    gemm_xw_kernel<<<blocks, 128, 0, stream>>>(x, W, xw, N);
  }
  // per-node logits: one wave per node
  {
    long long threads = (long long)N * 32;
    logits_kernel<<<(unsigned)((threads + 255) / 256), 256, 0, stream>>>(
        xw, att_src, att_dst, asrc, adst, N);
  }
  // softmax passes over edges
  edge_max_kernel<<<(tot + 255) / 256, 256, 0, stream>>>(ei, asrc, adst, mmax, E, N);
  edge_denom_kernel<<<(tot + 255) / 256, 256, 0, stream>>>(ei, asrc, adst, mmax, denom, E, N);
  // aggregation: one wave per edge
  {
    long long threads = (long long)tot * 32;
    edge_agg_kernel<<<(unsigned)((threads + 255) / 256), 256, 0, stream>>>(
        ei, xw, asrc, adst, mmax, denom, out, E, N);
  }
  // bias + ELU
  {
    int threads = N * DF;
    finalize_kernel<<<(threads + 255) / 256, 256, 0, stream>>>(out, bias, threads);
  }
}